// GAT_84593675862406
// MI455X (gfx1250) — compile-verified
//
#include <hip/hip_runtime.h>

typedef __attribute__((ext_vector_type(2))) float    v2f;
typedef __attribute__((ext_vector_type(4))) float    v4f;
typedef __attribute__((ext_vector_type(8))) float    v8f;
typedef __attribute__((ext_vector_type(4))) unsigned v4u;
typedef __attribute__((ext_vector_type(4))) int      v4i;
typedef __attribute__((ext_vector_type(8))) int      v8i;

#define N_NODES 512
#define HD      1024
#define BH      4096
#define NEDGE   16384
#define KBIG    524288   // N_NODES * HD

// ---- workspace layout (bytes) ----
#define WS_H       0u                               // 512*1024 f32 = 2 MB
#define WS_ATTN    (2u*1024u*1024u)                 // 512*512  f32 = 1 MB
#define WS_HPRIME  (WS_ATTN + 1u*1024u*1024u)      // 512*1024 f32 = 2 MB
#define WS_PACKE   (WS_HPRIME + 2u*1024u*1024u)    // 512*512  u64 = 2 MB
#define WS_PACKA   (WS_PACKE + 2u*1024u*1024u)     // 512*512  u64 = 2 MB
#define WS_S1      (WS_PACKA + 2u*1024u*1024u)     // 512 f32
#define WS_S2      (WS_S1 + 2048u)                  // 512 f32
#define WS_PART    (WS_S2 + 2048u)                  // 513*16 f32
#define WS_OUTRAW  (WS_PART + 513u*16u*4u)          // 513 f32
#define WS_SCAL    ((WS_OUTRAW + 513u*4u + 255u) & ~255u)  // 2 u32

__device__ __forceinline__ float unpack_val(unsigned long long k) {
  return (k >> 32) ? __uint_as_float((unsigned)(k & 0xFFFFFFFFull)) : 0.0f;
}

// ---- TDM: DMA a 2D f32 tile (tile_rows x tile_cols) from global into LDS ----
// D# per CDNA5 ISA: group0 = {flags, lds_addr, global_addr, type=2},
// group1 = {data_size=4B, tensor dims, tile dims, row stride}; remaining groups
// zero (tile_dim2..4 == 0 -> unused). Tracked by TENSORcnt.
// This toolchain exposes the 6-arg builtin: (v4u, v8i, v4i, v4i, v8i, i32 cpol).
__device__ __forceinline__ void tdm_load_tile_f32(const float* gsrc, void* lds_dst,
                                                  unsigned tile_cols, unsigned tile_rows,
                                                  unsigned tensor_cols, unsigned tensor_rows,
                                                  unsigned row_stride_elems) {
  const unsigned lds_addr = (unsigned)(uintptr_t)lds_dst;  // low 32b of flat = LDS offset
  const unsigned long long ga = (unsigned long long)(uintptr_t)gsrc;
  v4u g0;
  g0.x = 1u;                                               // count=1 (valid user D#)
  g0.y = lds_addr;                                         // bits 63:32 lds_addr
  g0.z = (unsigned)ga;                                     // global_addr lo
  g0.w = (unsigned)((ga >> 32) & 0x01FFFFFFu) | (2u << 30); // addr hi | type=2
  v8i g1;
  g1[0] = (int)(2u << 16);                                  // data_size = 4 bytes
  g1[1] = (int)((tensor_cols & 0xFFFFu) << 16);             // tensor_dim0 lo16
  g1[2] = (int)(((tensor_cols >> 16) & 0xFFFFu) |
                ((tensor_rows & 0xFFFFu) << 16));           // dim0 hi16 | dim1 lo16
  g1[3] = (int)(((tensor_rows >> 16) & 0xFFFFu) |
                ((tile_cols & 0xFFFFu) << 16));             // dim1 hi16 | tile_dim0
  g1[4] = (int)(tile_rows & 0xFFFFu);                       // tile_dim1 (tile_dim2=0)
  g1[5] = (int)row_stride_elems;                            // tensor_dim0_stride lo32
  g1[6] = 0;                                                // stride hi | dim1_stride
  g1[7] = 0;
  const v4i z4 = {0, 0, 0, 0};
  const v8i z8 = {0, 0, 0, 0, 0, 0, 0, 0};
  __builtin_amdgcn_tensor_load_to_lds(g0, g1, z4, z4, z8, 0);
}

// ---------------- zero packed scatter buffers + init minmax scalars ----------
__global__ void k_zero(unsigned long long* __restrict__ pe,
                       unsigned long long* __restrict__ pa,
                       unsigned* __restrict__ scal) {
  unsigned i = blockIdx.x * blockDim.x + threadIdx.x;
  if (i < (unsigned)(N_NODES * N_NODES)) { pe[i] = 0ull; pa[i] = 0ull; }
  if (i == 0) { scal[0] = 0x7F800000u; /* +inf bits (min) */ scal[1] = 0u; /* max */ }
}

// ---------------- h = final_x @ W  via V_WMMA_F32_16X16X4_F32 ----------------
// final_x[i, k] = X[(i&127)*4 + (k>>10), k&1023] + img[i>>7, k&1023]
// B tiles (64K x 128N of W) arrive via double-buffered TDM DMA.
__global__ __launch_bounds__(256) void k_gemm_h(const float* __restrict__ X,
                                                const float* __restrict__ img,
                                                const float* __restrict__ W,
                                                float* __restrict__ h) {
  __shared__ float lsA[16 * 64];
  __shared__ float lsB[2][64 * 128];
  const int tid  = threadIdx.x;
  const int wave = tid >> 5;
  const int lane = tid & 31;
  const int i0 = blockIdx.x * 16;    // M tile
  const int n0 = blockIdx.y * 128;   // N tile (8 waves x 16 cols)
  const int NIT = BH / 64;
  v8f acc0 = {}, acc1 = {};
  if (tid == 0)   // prologue DMA: first B tile
    tdm_load_tile_f32(W + n0, &lsB[0][0], 128, 64, HD, BH, HD);
  for (int it = 0; it < NIT; ++it) {
    const int kc  = it * 64;
    const int cur = it & 1;
    {   // A tile: 16 rows x 64 K of final_x (gather + add, 1 float4/thread)
      int idx = tid * 4;
      int rl  = idx >> 6;
      int tt  = idx & 63;
      int gi  = i0 + rl;
      int xrow = ((gi & 127) << 2) + (kc >> 10);
      int t0   = (kc & 1023) + tt;
      const float4 xv = *reinterpret_cast<const float4*>(X + xrow * HD + t0);
      const float4 iv = *reinterpret_cast<const float4*>(img + (gi >> 7) * HD + t0);
      float4 s; s.x = xv.x + iv.x; s.y = xv.y + iv.y; s.z = xv.z + iv.z; s.w = xv.w + iv.w;
      *reinterpret_cast<float4*>(&lsA[rl * 64 + tt]) = s;
    }
    if (tid == 0) {   // wave0: kick next DMA, then ensure current tile landed
      if (it + 1 < NIT) {
        tdm_load_tile_f32(W + (kc + 64) * HD + n0, &lsB[1 - cur][0], 128, 64, HD, BH, HD);
        __builtin_amdgcn_s_wait_tensorcnt((short)1);  // TDM in-order: current done
      } else {
        __builtin_amdgcn_s_wait_tensorcnt((short)0);
      }
    }
    __syncthreads();
    const float* Bt = &lsB[cur][0];
    const int ml   = lane & 15;
    const int koff = (lane >> 4) << 1;        // lanes 16-31 hold K+2,K+3
    const int cl   = (wave << 4) + ml;
#pragma unroll
    for (int kk = 0; kk < 64; kk += 8) {      // two independent accum chains
      v2f a, b;
      a.x = lsA[ml * 64 + kk + koff];
      a.y = lsA[ml * 64 + kk + koff + 1];
      b.x = Bt[(kk + koff) * 128 + cl];
      b.y = Bt[(kk + koff + 1) * 128 + cl];
      acc0 = __builtin_amdgcn_wmma_f32_16x16x4_f32(false, a, false, b,
                                                   (short)0, acc0, false, false);
      a.x = lsA[ml * 64 + kk + 4 + koff];
      a.y = lsA[ml * 64 + kk + 4 + koff + 1];
      b.x = Bt[(kk + 4 + koff) * 128 + cl];
      b.y = Bt[(kk + 5 + koff) * 128 + cl];
      acc1 = __builtin_amdgcn_wmma_f32_16x16x4_f32(false, a, false, b,
                                                   (short)0, acc1, false, false);
    }
    __syncthreads();
  }
  const int rbase = i0 + ((lane >> 4) << 3);  // lanes 16-31 hold rows M+8
  const int col   = n0 + (wave << 4) + (lane & 15);
#pragma unroll
  for (int v = 0; v < 8; ++v)
    h[(rbase + v) * HD + col] = acc0[v] + acc1[v];
}

// ---------------- h_prime = attn @ h  (same structure, K=512) ----------------
__global__ __launch_bounds__(256) void k_gemm_attn(const float* __restrict__ A,
                                                   const float* __restrict__ B,
                                                   float* __restrict__ C) {
  __shared__ float lsA[16 * 64];
  __shared__ float lsB[2][64 * 128];
  const int tid  = threadIdx.x;
  const int wave = tid >> 5;
  const int lane = tid & 31;
  const int i0 = blockIdx.x * 16;
  const int n0 = blockIdx.y * 128;
  const int NIT = N_NODES / 64;
  v8f acc0 = {}, acc1 = {};
  if (tid == 0)
    tdm_load_tile_f32(B + n0, &lsB[0][0], 128, 64, HD, N_NODES, HD);
  for (int it = 0; it < NIT; ++it) {
    const int kc  = it * 64;
    const int cur = it & 1;
    {
      int idx = tid * 4;
      int rl  = idx >> 6;
      int tt  = idx & 63;
      float4 av = *reinterpret_cast<const float4*>(A + (i0 + rl) * N_NODES + kc + tt);
      *reinterpret_cast<float4*>(&lsA[rl * 64 + tt]) = av;
    }
    if (tid == 0) {
      if (it + 1 < NIT) {
        tdm_load_tile_f32(B + (kc + 64) * HD + n0, &lsB[1 - cur][0], 128, 64, HD, N_NODES, HD);
        __builtin_amdgcn_s_wait_tensorcnt((short)1);
      } else {
        __builtin_amdgcn_s_wait_tensorcnt((short)0);
      }
    }
    __syncthreads();
    const float* Bt = &lsB[cur][0];
    const int ml   = lane & 15;
    const int koff = (lane >> 4) << 1;
    const int cl   = (wave << 4) + ml;
#pragma unroll
    for (int kk = 0; kk < 64; kk += 8) {
      v2f a, b;
      a.x = lsA[ml * 64 + kk + koff];
      a.y = lsA[ml * 64 + kk + koff + 1];
      b.x = Bt[(kk + koff) * 128 + cl];
      b.y = Bt[(kk + koff + 1) * 128 + cl];
      acc0 = __builtin_amdgcn_wmma_f32_16x16x4_f32(false, a, false, b,
                                                   (short)0, acc0, false, false);
      a.x = lsA[ml * 64 + kk + 4 + koff];
      a.y = lsA[ml * 64 + kk + 4 + koff + 1];
      b.x = Bt[(kk + 4 + koff) * 128 + cl];
      b.y = Bt[(kk + 5 + koff) * 128 + cl];
      acc1 = __builtin_amdgcn_wmma_f32_16x16x4_f32(false, a, false, b,
                                                   (short)0, acc1, false, false);
    }
    __syncthreads();
  }
  const int rbase = i0 + ((lane >> 4) << 3);
  const int col   = n0 + (wave << 4) + (lane & 15);
#pragma unroll
  for (int v = 0; v < 8; ++v)
    C[(rbase + v) * HD + col] = acc0[v] + acc1[v];
}

// ---------------- s1[n] = h[n]·a[0:H], s2[n] = h[n]·a[H:2H] ------------------
__global__ __launch_bounds__(256) void k_scores(const float* __restrict__ h,
                                                const float* __restrict__ a,
                                                float* __restrict__ s1,
                                                float* __restrict__ s2) {
  __shared__ float r1[256], r2[256];
  const int n = blockIdx.x, tid = threadIdx.x;
  float p1 = 0.f, p2 = 0.f;
  for (int j = tid; j < HD; j += 256) {
    float hv = h[n * HD + j];
    p1 += hv * a[j];
    p2 += hv * a[HD + j];
  }
  r1[tid] = p1; r2[tid] = p2; __syncthreads();
  for (int s = 128; s > 0; s >>= 1) {
    if (tid < s) { r1[tid] += r1[tid + s]; r2[tid] += r2[tid + s]; }
    __syncthreads();
  }
  if (tid == 0) { s1[n] = r1[0]; s2[n] = r2[0]; }
}

// ------- deterministic "last edge wins" scatter via u64 atomicMax ------------
__global__ void k_scatter(const int* __restrict__ er, const int* __restrict__ ec,
                          const float* __restrict__ av,
                          const float* __restrict__ s1, const float* __restrict__ s2,
                          unsigned long long* __restrict__ pe,
                          unsigned long long* __restrict__ pa) {
  const int i = blockIdx.x * 256 + threadIdx.x;
  const int r = er[i], c = ec[i];
  float ev = s1[r] + s2[c];
  ev = ev > 0.f ? ev : 0.2f * ev;                 // leaky_relu(0.2)
  const unsigned long long tag = ((unsigned long long)(i + 1)) << 32;
  atomicMax(&pe[r * N_NODES + c], tag | (unsigned long long)__float_as_uint(ev));
  atomicMax(&pa[r * N_NODES + c], tag | (unsigned long long)__float_as_uint(av[i]));
}

// ---------------- min/max over dense adj2 (values >= 0: uint trick) ----------
__global__ __launch_bounds__(256) void k_adjminmax(const unsigned long long* __restrict__ pa,
                                                   unsigned* __restrict__ scal) {
  __shared__ unsigned smin[256], smax[256];
  const int tid = threadIdx.x;
  const int base = (blockIdx.x * 256 + tid) * 4;
  float mn = 1e30f, mx = -1e30f;
#pragma unroll
  for (int j = 0; j < 4; ++j) {
    float v = 2.f * unpack_val(pa[base + j]);
    mn = fminf(mn, v); mx = fmaxf(mx, v);
  }
  smin[tid] = __float_as_uint(mn); smax[tid] = __float_as_uint(mx);
  __syncthreads();
  for (int s = 128; s > 0; s >>= 1) {
    if (tid < s) {
      smin[tid] = min(smin[tid], smin[tid + s]);
      smax[tid] = max(smax[tid], smax[tid + s]);
    }
    __syncthreads();
  }
  if (tid == 0) { atomicMin(&scal[0], smin[0]); atomicMax(&scal[1], smax[0]); }
}

// ------- row softmax over dense e (incl. structural zeros) + new_adj ---------
__global__ __launch_bounds__(256) void k_softmax_newadj(const unsigned long long* __restrict__ pe,
                                                        const unsigned long long* __restrict__ pa,
                                                        const unsigned* __restrict__ scal,
                                                        float* __restrict__ attn,
                                                        float* __restrict__ newadj) {
  __shared__ float red[256];
  const int row = blockIdx.x, tid = threadIdx.x;
  const float e0 = unpack_val(pe[row * N_NODES + tid]);
  const float e1 = unpack_val(pe[row * N_NODES + tid + 256]);
  red[tid] = fmaxf(e0, e1); __syncthreads();
  for (int s = 128; s > 0; s >>= 1) {
    if (tid < s) red[tid] = fmaxf(red[tid], red[tid + s]);
    __syncthreads();
  }
  const float rowmax = red[0]; __syncthreads();
  const float x0 = __expf(e0 - rowmax), x1 = __expf(e1 - rowmax);
  red[tid] = x0 + x1; __syncthreads();
  for (int s = 128; s > 0; s >>= 1) {
    if (tid < s) red[tid] += red[tid + s];
    __syncthreads();
  }
  const float inv = 1.0f / red[0];
  const float amin = __uint_as_float(scal[0]);
  const float amax = __uint_as_float(scal[1]);
  float den = amax - amin; den = (den == 0.f) ? 1.f : den;
  const float invden = 1.0f / den;
  const float a0 = x0 * inv, a1 = x1 * inv;
  attn[row * N_NODES + tid]       = a0;
  attn[row * N_NODES + tid + 256] = a1;
  const float v0 = 2.f * unpack_val(pa[row * N_NODES + tid]);
  const float v1 = 2.f * unpack_val(pa[row * N_NODES + tid + 256]);
  newadj[row * N_NODES + tid]       = a0 * (v0 - amin) * invden;
  newadj[row * N_NODES + tid + 256] = a1 * (v1 - amin) * invden;
}

// ------- bandwidth stage: out[m] = lin_w[m,:]·hp (1.076 GB NT stream) --------
__global__ __launch_bounds__(256) void k_matvec(const float* __restrict__ lw,
                                                const float* __restrict__ hp,
                                                float* __restrict__ part) {
  __shared__ float red[256];
  const int m = blockIdx.x;                 // 0..512
  const int s = blockIdx.y;                 // 0..15
  const int tid = threadIdx.x;
  const int CHUNK = KBIG / 16;              // 32768 floats = 128 KB
  const float* lrow = lw + (size_t)m * KBIG + (size_t)s * CHUNK;
  const float* hrow = hp + (size_t)s * CHUNK;
  float acc = 0.f;
#pragma unroll 4
  for (int j = 0; j < 32; ++j) {
    const int idx = (tid + j * 256) * 4;
    const v4f wv = __builtin_nontemporal_load(reinterpret_cast<const v4f*>(lrow + idx));
    const float4 hv = *reinterpret_cast<const float4*>(hrow + idx);
    acc += wv.x * hv.x + wv.y * hv.y + wv.z * hv.z + wv.w * hv.w;
  }
  red[tid] = acc; __syncthreads();
  for (int t = 128; t > 0; t >>= 1) {
    if (tid < t) red[tid] += red[tid + t];
    __syncthreads();
  }
  if (tid == 0) part[m * 16 + s] = red[0];
}

__global__ void k_vecreduce(const float* __restrict__ part,
                            const float* __restrict__ lb,
                            float* __restrict__ outraw) {
  const int m = blockIdx.x * blockDim.x + threadIdx.x;
  if (m < 513) {
    float s = lb[m];
#pragma unroll
    for (int j = 0; j < 16; ++j) s += part[m * 16 + j];   // fixed order: deterministic
    outraw[m] = s;
  }
}

__global__ __launch_bounds__(512) void k_final(const float* __restrict__ outraw,
                                               float* __restrict__ outn) {
  __shared__ float rmn[512], rmx[512];
  const int tid = threadIdx.x;
  float v = outraw[tid];
  float mn = v, mx = v;
  if (tid == 0) { float w = outraw[512]; mn = fminf(mn, w); mx = fmaxf(mx, w); }
  rmn[tid] = mn; rmx[tid] = mx; __syncthreads();
  for (int s = 256; s > 0; s >>= 1) {
    if (tid < s) {
      rmn[tid] = fminf(rmn[tid], rmn[tid + s]);
      rmx[tid] = fmaxf(rmx[tid], rmx[tid + s]);
    }
    __syncthreads();
  }
  const float gmn = rmn[0], gmx = rmx[0];
  const float den = gmx - gmn;
  if (den == 0.f) {
    outn[tid] = 0.5f;
    if (tid == 0) outn[512] = 0.5f;
  } else {
    const float inv = 1.f / den;
    outn[tid] = (outraw[tid] - gmn) * inv;
    if (tid == 0) outn[512] = (outraw[512] - gmn) * inv;
  }
}

extern "C" void kernel_launch(void* const* d_in, const int* in_sizes, int n_in,
                              void* d_out, int out_size, void* d_ws, size_t ws_size,
                              hipStream_t stream) {
  (void)in_sizes; (void)n_in; (void)out_size; (void)ws_size;
  const float* img = (const float*)d_in[0];
  const int*   er  = (const int*)d_in[1];
  const int*   ec  = (const int*)d_in[2];
  const float* av  = (const float*)d_in[3];
  const float* X   = (const float*)d_in[4];
  const float* W   = (const float*)d_in[5];
  const float* a   = (const float*)d_in[6];
  const float* lw  = (const float*)d_in[7];
  const float* lb  = (const float*)d_in[8];

  char* ws = (char*)d_ws;
  float* h      = (float*)(ws + WS_H);
  float* attn   = (float*)(ws + WS_ATTN);
  float* hp     = (float*)(ws + WS_HPRIME);
  unsigned long long* pe = (unsigned long long*)(ws + WS_PACKE);
  unsigned long long* pa = (unsigned long long*)(ws + WS_PACKA);
  float* s1     = (float*)(ws + WS_S1);
  float* s2     = (float*)(ws + WS_S2);
  float* part   = (float*)(ws + WS_PART);
  float* outraw = (float*)(ws + WS_OUTRAW);
  unsigned* scal = (unsigned*)(ws + WS_SCAL);

  float* out_adj  = (float*)d_out;                 // 512*512
  float* out_norm = out_adj + N_NODES * N_NODES;   // 513

  k_zero<<<512, 512, 0, stream>>>(pe, pa, scal);
  k_gemm_h<<<dim3(32, 8), 256, 0, stream>>>(X, img, W, h);
  k_scores<<<512, 256, 0, stream>>>(h, a, s1, s2);
  k_scatter<<<NEDGE / 256, 256, 0, stream>>>(er, ec, av, s1, s2, pe, pa);
  k_adjminmax<<<256, 256, 0, stream>>>(pa, scal);
  k_softmax_newadj<<<512, 256, 0, stream>>>(pe, pa, scal, attn, out_adj);
  k_gemm_attn<<<dim3(32, 8), 256, 0, stream>>>(attn, h, hp);
  k_matvec<<<dim3(513, 16), 256, 0, stream>>>(lw, hp, part);
  k_vecreduce<<<3, 256, 0, stream>>>(part, lb, outraw);
  k_final<<<1, 512, 0, stream>>>(outraw, out_norm);
}